// OptPosEncBatch_51281909514407
// MI455X (gfx1250) — compile-verified
//
#include <hip/hip_runtime.h>
#include <hip/hip_bf16.h>

#ifndef __has_builtin
#define __has_builtin(x) 0
#endif

// Problem constants (from reference)
#define CODE_NUM   64
#define NFEAT      3
#define NCHAN      64
#define NBATCH     8
#define NPTS       65536
#define NCOLS      (CODE_NUM * NFEAT)          // 192 columns per shape slice
#define ROWSTRIDE  (16 * NFEAT * CODE_NUM)     // 3072 columns total per channel row
// Tuning
#define LDS_STRIDE 68                          // 192-col tile: 68 % 64 == 4 -> only 2-way
                                               // conflicts on staging writes, none on reads
#define PTS_PER_WG 1024
#define NTHREADS   256
#define CHUNKS_PER_B (NPTS / PTS_PER_WG)       // 64

// gfx1250 async global->LDS staging path (ASYNCcnt). Falls back to a plain
// synchronous copy if the builtins are not available on this toolchain.
#if defined(__HIP_DEVICE_COMPILE__) && defined(__gfx1250__) && \
    __has_builtin(__builtin_amdgcn_global_load_async_to_lds_b32) && \
    __has_builtin(__builtin_amdgcn_s_wait_asynccnt)
#define USE_ASYNC_LDS 1
typedef __attribute__((address_space(1))) int g1_int;
typedef __attribute__((address_space(3))) int l3_int;
#else
#define USE_ASYNC_LDS 0
#endif

__device__ __forceinline__ void acc4(float4& acc, float wt, const float* p)
{
    const float4 v = *(const float4*)p;
    acc.x = fmaf(wt, v.x, acc.x);
    acc.y = fmaf(wt, v.y, acc.y);
    acc.z = fmaf(wt, v.z, acc.z);
    acc.w = fmaf(wt, v.w, acc.w);
}

__global__ __launch_bounds__(NTHREADS)
void OptPosEncBatch_51281909514407_kernel(const float* __restrict__ coords,
                                          const int*   __restrict__ shape_idx,
                                          const float* __restrict__ shape_code,
                                          float*       __restrict__ out)
{
    // Codebook slice for this batch, transposed to [col][chan(+pad)] so the
    // gather phase reads consecutive channels across lanes (conflict-free,
    // 16B-aligned float4 ds loads).  192 * 68 * 4 = 52,224 B of the 320 KB LDS.
    __shared__ __align__(16) float tile[NCOLS * LDS_STRIDE];

    const int tid   = threadIdx.x;
    const int blk   = blockIdx.x;
    const int b     = blk / CHUNKS_PER_B;
    const int chunk = blk % CHUNKS_PER_B;
    const int base  = shape_idx[b] * NCOLS;    // column offset of this shape's slice

    // ---- Stage 64x192 slice (48 KB) into LDS, transposed ----
    const float* src = shape_code + base;
    #pragma unroll
    for (int i = 0; i < (NCHAN * NCOLS) / NTHREADS; ++i) {   // 48 iterations
        const int j   = i * NTHREADS + tid;
        const int c   = j / NCOLS;            // channel row (global reads coalesced per row)
        const int col = j - c * NCOLS;
#if USE_ASYNC_LDS
        __builtin_amdgcn_global_load_async_to_lds_b32(
            (g1_int*)(src + (size_t)c * ROWSTRIDE + col),
            (l3_int*)(&tile[col * LDS_STRIDE + c]),
            0, 0);
#else
        tile[col * LDS_STRIDE + c] = src[(size_t)c * ROWSTRIDE + col];
#endif
    }
#if USE_ASYNC_LDS
    __builtin_amdgcn_s_wait_asynccnt(0);
#endif
    __syncthreads();

    // ---- Gather + interpolate ----
    const int lane = tid & 31;
    const int wave = tid >> 5;
    const int grp  = lane >> 4;   // which of 2 points this half-wave handles
    const int m    = lane & 15;   // channel quad: channels 4m..4m+3

    const float* tl    = tile + 4 * m;
    const long   pbase = (long)b * NPTS + (long)chunk * PTS_PER_WG;

    for (int it = 0; it < PTS_PER_WG / 16; ++it) {   // 8 waves * 2 pts = 16 pts / iter
        const int  p_local = it * 16 + wave * 2 + grp;
        const long gp      = pbase + p_local;

        const float* cp = coords + gp * 3;           // 16 lanes share -> L0 broadcast
        const float g0 = (cp[0] + 1.0f) * ((CODE_NUM - 1) * 0.5f);
        const float g1 = (cp[1] + 1.0f) * ((CODE_NUM - 1) * 0.5f);
        const float g2 = (cp[2] + 1.0f) * ((CODE_NUM - 1) * 0.5f);
        const int i0 = (int)g0;                      // g >= 0, trunc == floor
        const int i1 = (int)g1;
        const int i2 = (int)g2;
        const float f0 = g0 - (float)i0;
        const float f1 = g1 - (float)i1;
        const float f2 = g2 - (float)i2;
        const int c0 = i0;
        const int c1 = CODE_NUM + i1;
        const int c2 = 2 * CODE_NUM + i2;

        float4 acc = {0.0f, 0.0f, 0.0f, 0.0f};
        acc4(acc, 1.0f - f0, tl + (c0    ) * LDS_STRIDE);
        acc4(acc, f0,        tl + (c0 + 1) * LDS_STRIDE);
        acc4(acc, 1.0f - f1, tl + (c1    ) * LDS_STRIDE);
        acc4(acc, f1,        tl + (c1 + 1) * LDS_STRIDE);
        acc4(acc, 1.0f - f2, tl + (c2    ) * LDS_STRIDE);
        acc4(acc, f2,        tl + (c2 + 1) * LDS_STRIDE);

        // 16 lanes -> 256 B contiguous per point, adjacent points adjacent.
        *(float4*)(out + gp * NCHAN + 4 * m) = acc;
    }
}

extern "C" void kernel_launch(void* const* d_in, const int* in_sizes, int n_in,
                              void* d_out, int out_size, void* d_ws, size_t ws_size,
                              hipStream_t stream) {
    const float* coords     = (const float*)d_in[0];
    const int*   idx        = (const int*)  d_in[1];
    const float* shape_code = (const float*)d_in[2];
    float*       out        = (float*)d_out;

    dim3 grid(NBATCH * CHUNKS_PER_B);   // 512 workgroups
    dim3 block(NTHREADS);               // 8 wave32 waves
    OptPosEncBatch_51281909514407_kernel<<<grid, block, 0, stream>>>(
        coords, idx, shape_code, out);
}